// MultiBoxLoss_15401752723760
// MI455X (gfx1250) — compile-verified
//
#include <hip/hip_runtime.h>

#define THRESH 0.105f   // max(0.05, 0.35*0.3)
#define VAR0   0.1f
#define VAR1   0.2f

typedef __attribute__((ext_vector_type(2))) float v2f;
typedef __attribute__((ext_vector_type(8))) float v8f;

__device__ __forceinline__ float sm_l1(float x) {
    float ax = fabsf(x);
    return ax < 1.0f ? 0.5f * x * x : ax - 0.5f;
}

// ---------------- Phase 0: zero ws accumulators -------------------------
__global__ void k_zero(unsigned long long* keys, int nk, int* numPos, int B, int* counters) {
    int i = blockIdx.x * blockDim.x + threadIdx.x;
    if (i < nk) keys[i] = 0ull;
    if (i < B)  numPos[i] = 0;
    if (i < 2)  counters[i] = 0;
}

// ---------------- Phase A: matching (per-prior max, per-truth argmax) ---
__global__ void k_match(const float* __restrict__ priors,
                        const float* __restrict__ targets,
                        float* __restrict__ bto, int* __restrict__ bti,
                        unsigned long long* __restrict__ keys,
                        int P, int G) {
    int b = blockIdx.y;
    int j = blockIdx.x * blockDim.x + threadIdx.x;
    __shared__ float tr[64 * 5];                 // x0,y0,x1,y1,area per truth
    __shared__ unsigned long long lkeys[64];
    for (int g = threadIdx.x; g < G; g += blockDim.x) {
        const float* t = targets + ((size_t)b * G + g) * 15;
        float x0 = t[0], y0 = t[1], x1 = t[2], y1 = t[3];
        tr[g * 5 + 0] = x0; tr[g * 5 + 1] = y0;
        tr[g * 5 + 2] = x1; tr[g * 5 + 3] = y1;
        tr[g * 5 + 4] = (x1 - x0) * (y1 - y0);
        lkeys[g] = 0ull;
    }
    __syncthreads();
    if (j < P) {
        float px0 = priors[j * 4 + 0], py0 = priors[j * 4 + 1];
        float px1 = priors[j * 4 + 2], py1 = priors[j * 4 + 3];
        float pa = (px1 - px0) * (py1 - py0);
        float bestOv = -1.0f; int bestG = 0;
        for (int g = 0; g < G; ++g) {
            float ix0 = fmaxf(tr[g * 5 + 0], px0), iy0 = fmaxf(tr[g * 5 + 1], py0);
            float ix1 = fminf(tr[g * 5 + 2], px1), iy1 = fminf(tr[g * 5 + 3], py1);
            float iw = fmaxf(ix1 - ix0, 0.0f), ih = fmaxf(iy1 - iy0, 0.0f);
            float inter = iw * ih;
            float iou = inter / (tr[g * 5 + 4] + pa - inter);
            if (iou > bestOv) { bestOv = iou; bestG = g; }   // first-max tie rule
            unsigned long long key =
                ((unsigned long long)__float_as_uint(iou) << 32) | (unsigned int)j;
            atomicMax(&lkeys[g], key);
        }
        bto[(size_t)b * P + j] = bestOv;
        bti[(size_t)b * P + j] = bestG;
    }
    __syncthreads();
    for (int g = threadIdx.x; g < G; g += blockDim.x)
        atomicMax(&keys[(size_t)b * G + g], lkeys[g]);
}

// ---------------- Phase B: force-match override (last g wins) -----------
__global__ void k_override(const unsigned long long* __restrict__ keys,
                           float* __restrict__ bto, int* __restrict__ bti,
                           int P, int G) {
    int b = blockIdx.x;
    if (threadIdx.x == 0) {
        for (int g = 0; g < G; ++g) {
            unsigned long long key = keys[(size_t)b * G + g];
            int j = (int)(key & 0xFFFFFFFFull);
            bto[(size_t)b * P + j] = 2.0f;
            bti[(size_t)b * P + j] = g;
        }
    }
}

// ---------------- Phase C: encode + smooth-L1 + CE, block partials ------
__global__ void k_loss(const float* __restrict__ loc_data,
                       const float* __restrict__ conf_data,
                       const float* __restrict__ landm_data,
                       const float* __restrict__ priors,
                       const float* __restrict__ targets,
                       const float* __restrict__ bto, const int* __restrict__ bti,
                       float* __restrict__ loss_c, float* __restrict__ partials,
                       int* __restrict__ numPos, int* __restrict__ counters,
                       int P, int G) {
    int b = blockIdx.y;
    int j = blockIdx.x * blockDim.x + threadIdx.x;
    int tid = threadIdx.x;
    float sl = 0.0f, slm = 0.0f, scp = 0.0f;
    int cp = 0, cp1 = 0;
    if (j < P) {
        float over = bto[(size_t)b * P + j];
        int g = bti[(size_t)b * P + j];
        const float* t = targets + ((size_t)b * G + g) * 15;
        int conf = (int)t[14];
        if (over < THRESH) conf = 0;
        bool pos = (conf != 0), pos1 = (conf > 0);
        float px0 = priors[j * 4 + 0], py0 = priors[j * 4 + 1];
        float px1 = priors[j * 4 + 2], py1 = priors[j * 4 + 3];
        float pcx = 0.5f * (px0 + px1), pcy = 0.5f * (py0 + py1);
        float pw = px1 - px0, ph = py1 - py0;
        if (pos) {
            float gx = (0.5f * (t[0] + t[2]) - pcx) / (VAR0 * pw);
            float gy = (0.5f * (t[1] + t[3]) - pcy) / (VAR0 * ph);
            float gw = logf((t[2] - t[0]) / pw) / VAR1;
            float gh = logf((t[3] - t[1]) / ph) / VAR1;
            const float* ld = loc_data + ((size_t)b * P + j) * 4;
            sl = sm_l1(ld[0] - gx) + sm_l1(ld[1] - gy) +
                 sm_l1(ld[2] - gw) + sm_l1(ld[3] - gh);
            cp = 1;
        }
        if (pos1) {
            const float* lm = landm_data + ((size_t)b * P + j) * 10;
            #pragma unroll
            for (int i = 0; i < 5; ++i) {
                float gx = (t[4 + 2 * i] - pcx) / (VAR0 * pw);
                float gy = (t[5 + 2 * i] - pcy) / (VAR0 * ph);
                slm += sm_l1(lm[2 * i] - gx) + sm_l1(lm[2 * i + 1] - gy);
            }
            cp1 = 1;
        }
        const float* cd = conf_data + ((size_t)b * P + j) * 2;
        float c0 = cd[0], c1 = cd[1];
        float m = fmaxf(c0, c1);
        float logZ = m + logf(expf(c0 - m) + expf(c1 - m));
        float ce = logZ - (pos ? c1 : c0);
        loss_c[(size_t)b * P + j] = pos ? 0.0f : ce;
        if (pos) scp = ce;
    }
    __shared__ float red[256];
    __shared__ int ired[256];
    // loc partial
    red[tid] = sl; __syncthreads();
    for (int s = 128; s > 0; s >>= 1) { if (tid < s) red[tid] += red[tid + s]; __syncthreads(); }
    float totL = red[0]; __syncthreads();
    // landm partial
    red[tid] = slm; __syncthreads();
    for (int s = 128; s > 0; s >>= 1) { if (tid < s) red[tid] += red[tid + s]; __syncthreads(); }
    float totLM = red[0]; __syncthreads();
    // ce-over-pos partial
    red[tid] = scp; __syncthreads();
    for (int s = 128; s > 0; s >>= 1) { if (tid < s) red[tid] += red[tid + s]; __syncthreads(); }
    float totCP = red[0]; __syncthreads();
    // counts
    ired[tid] = cp; __syncthreads();
    for (int s = 128; s > 0; s >>= 1) { if (tid < s) ired[tid] += ired[tid + s]; __syncthreads(); }
    int cpTot = ired[0]; __syncthreads();
    ired[tid] = cp1; __syncthreads();
    for (int s = 128; s > 0; s >>= 1) { if (tid < s) ired[tid] += ired[tid + s]; __syncthreads(); }
    int cp1Tot = ired[0];
    if (tid == 0) {
        int blockId = blockIdx.y * gridDim.x + blockIdx.x;
        partials[blockId * 3 + 0] = totL;
        partials[blockId * 3 + 1] = totLM;
        partials[blockId * 3 + 2] = totCP;
        atomicAdd(&numPos[b], cpTot);
        atomicAdd(&counters[0], cpTot);
        atomicAdd(&counters[1], cp1Tot);
    }
}

// ---------------- Phase D: per-batch radix select (top-K sum) -----------
__global__ void k_select(const float* __restrict__ loss_c,
                         const int* __restrict__ numPos,
                         float* __restrict__ negSum, int P) {
    int b = blockIdx.x;
    int tid = threadIdx.x;
    __shared__ unsigned int hist[256];
    __shared__ unsigned int s_prefix;
    __shared__ int s_need;
    __shared__ float red[256];
    int K = 3 * numPos[b];
    int Kc = (P - 1 < K) ? (P - 1) : K;
    if (Kc <= 0) { if (tid == 0) negSum[b] = 0.0f; return; }
    if (tid == 0) { s_prefix = 0u; s_need = Kc; }
    __syncthreads();
    const float* lc = loss_c + (size_t)b * P;
    for (int r = 0; r < 4; ++r) {
        int shift = 24 - 8 * r;
        hist[tid] = 0u;
        __syncthreads();
        unsigned int highmask = (unsigned int)(0xFFFFFFFFull << (shift + 8));
        unsigned int pfx = s_prefix;
        for (int j = tid; j < P; j += blockDim.x) {
            unsigned int u = __float_as_uint(lc[j]);   // ce >= 0: uint order == float order
            if ((u & highmask) == (pfx & highmask))
                atomicAdd(&hist[(u >> shift) & 255u], 1u);
        }
        __syncthreads();
        if (tid == 0) {
            int need = s_need;
            unsigned int pfx2 = s_prefix;
            for (int bin = 255; bin >= 0; --bin) {
                int h = (int)hist[bin];
                if (h >= need) {
                    s_prefix = pfx2 | ((unsigned int)bin << shift);
                    s_need = need;        // ties still needed at this level
                    break;
                }
                need -= h;                // these are strictly greater than T
            }
        }
        __syncthreads();
    }
    unsigned int Tbits = s_prefix;
    int needF = s_need;                   // == K - cntGreater
    float sum = 0.0f;
    for (int j = tid; j < P; j += blockDim.x) {
        float v = lc[j];
        if (__float_as_uint(v) > Tbits) sum += v;
    }
    red[tid] = sum; __syncthreads();
    for (int s = 128; s > 0; s >>= 1) { if (tid < s) red[tid] += red[tid + s]; __syncthreads(); }
    if (tid == 0)
        negSum[b] = red[0] + (float)needF * __uint_as_float(Tbits);
}

// ---------------- Phase E: WMMA-based final reduction -------------------
// One wave. Reduce 32 per-lane f32 accumulators with V_WMMA_F32_16X16X4_F32:
// A[m][0]=lane m, A[m][2]=lane m+16 (VGPR1 zero), B=ones => D[m][*] = v_m + v_{m+16}.
__device__ __forceinline__ float wave_reduce_wmma(float v) {
    v2f a; a.x = v; a.y = 0.0f;
    v2f bm; bm.x = 1.0f; bm.y = 1.0f;
    v8f c = {};
    c = __builtin_amdgcn_wmma_f32_16x16x4_f32(false, a, false, bm,
                                              (short)0, c, false, false);
    float s = 0.0f;
    #pragma unroll
    for (int r = 0; r < 8; ++r) {
        s += __int_as_float(__builtin_amdgcn_readlane(__float_as_int(c[r]), 0));
        s += __int_as_float(__builtin_amdgcn_readlane(__float_as_int(c[r]), 16));
    }
    return s;
}

__global__ void k_final(const float* __restrict__ partials, int nPart,
                        const float* __restrict__ negSum,
                        const int* __restrict__ counters,
                        float* __restrict__ out, int B) {
    int lane = threadIdx.x;
    float accL = 0.0f, accLM = 0.0f, accCP = 0.0f;
    for (int i = lane; i < nPart; i += 32) {
        accL  += partials[i * 3 + 0];
        accLM += partials[i * 3 + 1];
        accCP += partials[i * 3 + 2];
    }
    float accNeg = 0.0f;
    for (int i = lane; i < B; i += 32) accNeg += negSum[i];
    float SL   = wave_reduce_wmma(accL);
    float SLM  = wave_reduce_wmma(accLM);
    float SCP  = wave_reduce_wmma(accCP);
    float SNEG = wave_reduce_wmma(accNeg);
    if (lane == 0) {
        float N  = fmaxf((float)counters[0], 1.0f);
        float N1 = fmaxf((float)counters[1], 1.0f);
        out[0] = SL / N;
        out[1] = (SCP + SNEG) / N;
        out[2] = SLM / N1;
    }
}

// ---------------- Host launch -------------------------------------------
extern "C" void kernel_launch(void* const* d_in, const int* in_sizes, int n_in,
                              void* d_out, int out_size, void* d_ws, size_t ws_size,
                              hipStream_t stream) {
    const float* loc_data   = (const float*)d_in[0];
    const float* conf_data  = (const float*)d_in[1];
    const float* landm_data = (const float*)d_in[2];
    const float* priors     = (const float*)d_in[3];
    const float* targets    = (const float*)d_in[4];
    float* out = (float*)d_out;

    const int P = in_sizes[3] / 4;
    const int B = in_sizes[0] / (P * 4);
    const int G = in_sizes[4] / (B * 15);
    const int chunks = (P + 255) / 256;
    const int NB = chunks * B;

    char* ws = (char*)d_ws;
    float* bto = (float*)ws;                      ws += (size_t)B * P * sizeof(float);
    float* loss_c = (float*)ws;                   ws += (size_t)B * P * sizeof(float);
    int* bti = (int*)ws;                          ws += (size_t)B * P * sizeof(int);
    unsigned long long* keys = (unsigned long long*)ws; ws += (size_t)B * G * sizeof(unsigned long long);
    int* numPos = (int*)ws;                       ws += (size_t)B * sizeof(int);
    float* negSum = (float*)ws;                   ws += (size_t)B * sizeof(float);
    float* partials = (float*)ws;                 ws += (size_t)NB * 3 * sizeof(float);
    int* counters = (int*)ws;                     ws += 2 * sizeof(int);

    int nzero = B * G;                            // keys dominate; also covers B and 2
    k_zero<<<(nzero + 255) / 256, 256, 0, stream>>>(keys, nzero, numPos, B, counters);
    k_match<<<dim3(chunks, B), 256, 0, stream>>>(priors, targets, bto, bti, keys, P, G);
    k_override<<<B, 32, 0, stream>>>(keys, bto, bti, P, G);
    k_loss<<<dim3(chunks, B), 256, 0, stream>>>(loc_data, conf_data, landm_data, priors,
                                                targets, bto, bti, loss_c, partials,
                                                numPos, counters, P, G);
    k_select<<<B, 256, 0, stream>>>(loss_c, numPos, negSum, P);
    k_final<<<1, 32, 0, stream>>>(partials, NB, negSum, counters, out, B);
}